// Model_43593918055127
// MI455X (gfx1250) — compile-verified
//
#include <hip/hip_runtime.h>
#include <hip/hip_bf16.h>
#include <math.h>

// ---------------- problem constants (match reference) ----------------
constexpr int B_  = 16;
constexpr int L_  = 2048;
constexpr int C_  = 512;
constexpr int H_  = 4;
constexpr int E_  = 16384;
constexpr int NHE = 512;
constexpr int DH  = C_ / H_;      // 128
constexpr float NEG_SLOPE = 0.2f;

typedef float v2f __attribute__((ext_vector_type(2)));
typedef float v8f __attribute__((ext_vector_type(8)));

// ---------------- helpers ----------------
// monotonic float <-> uint transform so atomicMax(u32) implements float max
__device__ __forceinline__ unsigned f2o(float f) {
    unsigned u = __float_as_uint(f);
    return (u & 0x80000000u) ? ~u : (u | 0x80000000u);
}
__device__ __forceinline__ float o2f(unsigned u) {
    return (u & 0x80000000u) ? __uint_as_float(u & 0x7FFFFFFFu)
                             : __uint_as_float(~u);
}

// ---------------- init kernels ----------------
__global__ void k_zero_f32(float* p, long long n) {
    long long i = (long long)blockIdx.x * blockDim.x + threadIdx.x;
    long long stride = (long long)gridDim.x * blockDim.x;
    for (; i < n; i += stride) p[i] = 0.0f;
}

__global__ void k_init_max(unsigned* p, long long n) {
    long long i = (long long)blockIdx.x * blockDim.x + threadIdx.x;
    long long stride = (long long)gridDim.x * blockDim.x;
    const unsigned NEG_INF_O = 0x007FFFFFu; // f2o(-inf)
    for (; i < n; i += stride) p[i] = NEG_INF_O;
}

// ---------------- GEMM: xw[b,l,c] = sum_k x[b,l,k] * W[k,c]  (f32 WMMA) ----
// block = 256 threads = 8 waves; tile BM=128 x BN=64, K-chunk BK=32.
// wave w computes rows [w*16, w*16+16) of the tile, all 4 N-subtiles.
constexpr int BM = 128, BN = 64, BK = 32;

__global__ __launch_bounds__(256)
void k_gemm_wmma(const float* __restrict__ x,   // [B,L,C]
                 const float* __restrict__ W,   // [C,C]
                 float* __restrict__ xw)        // [B,L,C]
{
    __shared__ float As[BM][BK + 1];  // +1 pad: conflict-free column reads
    __shared__ float Bs[BK][BN + 1];

    const int b  = blockIdx.z;
    const int M0 = blockIdx.y * BM;
    const int N0 = blockIdx.x * BN;

    const int tid  = threadIdx.x;
    const int wave = tid >> 5;
    const int lane = tid & 31;
    const int l16  = lane & 15;       // lane-in-halfwave
    const int hsel = lane >> 4;       // 0: lanes 0-15, 1: lanes 16-31
    const int koff = hsel * 2;        // A/B fragment K offset per ISA layout

    v8f acc0 = {}, acc1 = {}, acc2 = {}, acc3 = {};

    const float* xrow = x + (size_t)(b * L_) * C_;

    for (int kt = 0; kt < C_; kt += BK) {
        // ---- stage A tile: 128x32 floats = 1024 float4, 4 per thread ----
        #pragma unroll
        for (int i = 0; i < 4; ++i) {
            int flat = tid + i * 256;          // float4 index
            int row  = flat >> 3;              // 8 float4 per row
            int kq   = flat & 7;
            const float4 v = *reinterpret_cast<const float4*>(
                xrow + (size_t)(M0 + row) * C_ + kt + kq * 4);
            As[row][kq * 4 + 0] = v.x;
            As[row][kq * 4 + 1] = v.y;
            As[row][kq * 4 + 2] = v.z;
            As[row][kq * 4 + 3] = v.w;
        }
        // ---- stage B tile: 32x64 floats = 512 float4, 2 per thread ----
        #pragma unroll
        for (int i = 0; i < 2; ++i) {
            int flat = tid + i * 256;
            int krow = flat >> 4;              // 16 float4 per row
            int nq   = flat & 15;
            const float4 v = *reinterpret_cast<const float4*>(
                W + (size_t)(kt + krow) * C_ + N0 + nq * 4);
            Bs[krow][nq * 4 + 0] = v.x;
            Bs[krow][nq * 4 + 1] = v.y;
            Bs[krow][nq * 4 + 2] = v.z;
            Bs[krow][nq * 4 + 3] = v.w;
        }
        __syncthreads();

        const int mrow = wave * 16 + l16;
        #pragma unroll
        for (int kk = 0; kk < BK; kk += 4) {
            // A fragment (16x4): lanes 0-15 hold K=kk..kk+1, lanes 16-31 K=kk+2..kk+3
            v2f a;
            a.x = As[mrow][kk + koff + 0];
            a.y = As[mrow][kk + koff + 1];
            // B fragments (4x16) for the 4 N-subtiles; same K split across halves
            v2f b0, b1, b2, b3;
            b0.x = Bs[kk + koff + 0][ 0 + l16];  b0.y = Bs[kk + koff + 1][ 0 + l16];
            b1.x = Bs[kk + koff + 0][16 + l16];  b1.y = Bs[kk + koff + 1][16 + l16];
            b2.x = Bs[kk + koff + 0][32 + l16];  b2.y = Bs[kk + koff + 1][32 + l16];
            b3.x = Bs[kk + koff + 0][48 + l16];  b3.y = Bs[kk + koff + 1][48 + l16];

            acc0 = __builtin_amdgcn_wmma_f32_16x16x4_f32(false, a, false, b0, (short)0, acc0, false, false);
            acc1 = __builtin_amdgcn_wmma_f32_16x16x4_f32(false, a, false, b1, (short)0, acc1, false, false);
            acc2 = __builtin_amdgcn_wmma_f32_16x16x4_f32(false, a, false, b2, (short)0, acc2, false, false);
            acc3 = __builtin_amdgcn_wmma_f32_16x16x4_f32(false, a, false, b3, (short)0, acc3, false, false);
        }
        __syncthreads();
    }

    // ---- write D: VGPR r -> M = r (lanes 0-15) / M = r+8 (lanes 16-31) ----
    float* out = xw + (size_t)(b * L_) * C_;
    const int rowBase = M0 + wave * 16 + hsel * 8;
    #pragma unroll
    for (int r = 0; r < 8; ++r) {
        const size_t base = (size_t)(rowBase + r) * C_ + N0 + l16;
        out[base +  0] = acc0[r];
        out[base + 16] = acc1[r];
        out[base + 32] = acc2[r];
        out[base + 48] = acc3[r];
    }
}

// ---------------- degree counts ----------------
__global__ void k_degree(const int* __restrict__ hidx,  // [B,2,E]
                         float* __restrict__ Dn,        // [B,L]
                         float* __restrict__ Bn)        // [B,NHE]
{
    int gid = blockIdx.x * blockDim.x + threadIdx.x;    // B*E
    if (gid >= B_ * E_) return;
    int e = gid % E_, b = gid / E_;
    int n0 = hidx[(b * 2 + 0) * E_ + e];
    int n1 = hidx[(b * 2 + 1) * E_ + e];
    atomicAdd(&Dn[b * L_ + n0], 1.0f);
    atomicAdd(&Bn[b * NHE + n1], 1.0f);
}

// ---------------- attention logits + leaky-relu + segment max ------------
__global__ void k_logits(const float* __restrict__ xw,   // [B,L,H,DH]
                         const int*   __restrict__ hidx, // [B,2,E]
                         const float* __restrict__ att,  // [1,H,2*DH]
                         float* __restrict__ logits,     // [B,E,H]
                         unsigned* __restrict__ maxb)    // [B,L,H]
{
    int gid = blockIdx.x * blockDim.x + threadIdx.x;     // B*E*H
    if (gid >= B_ * E_ * H_) return;
    int h = gid % H_, e = (gid / H_) % E_, b = gid / (H_ * E_);
    int n0 = hidx[(b * 2 + 0) * E_ + e];
    int n1 = hidx[(b * 2 + 1) * E_ + e];

    const float4* xi = reinterpret_cast<const float4*>(xw + (size_t)(b * L_ + n0) * C_ + h * DH);
    const float4* xj = reinterpret_cast<const float4*>(xw + (size_t)(b * L_ + n1) * C_ + h * DH);
    const float4* ai = reinterpret_cast<const float4*>(att + h * 2 * DH);
    const float4* aj = reinterpret_cast<const float4*>(att + h * 2 * DH + DH);

    float s = 0.0f;
    #pragma unroll 8
    for (int q = 0; q < DH / 4; ++q) {
        float4 xv = xi[q], av = ai[q];
        s += xv.x * av.x + xv.y * av.y + xv.z * av.z + xv.w * av.w;
        xv = xj[q]; av = aj[q];
        s += xv.x * av.x + xv.y * av.y + xv.z * av.z + xv.w * av.w;
    }
    s = (s > 0.0f) ? s : NEG_SLOPE * s;
    logits[gid] = s;
    atomicMax(&maxb[(b * L_ + n0) * H_ + h], f2o(s));
}

// ---------------- exp(a - max) + segment sum ----------------
__global__ void k_expsum(const int* __restrict__ hidx,
                         float* __restrict__ logits,      // in: a, out: exp(a-m)
                         const unsigned* __restrict__ maxb,
                         float* __restrict__ sumb)        // [B,L,H]
{
    int gid = blockIdx.x * blockDim.x + threadIdx.x;      // B*E*H
    if (gid >= B_ * E_ * H_) return;
    int h = gid % H_, e = (gid / H_) % E_, b = gid / (H_ * E_);
    int n0 = hidx[(b * 2 + 0) * E_ + e];
    float m = o2f(maxb[(b * L_ + n0) * H_ + h]);
    if (!isfinite(m)) m = 0.0f;
    float ev = expf(logits[gid] - m);
    logits[gid] = ev;
    atomicAdd(&sumb[(b * L_ + n0) * H_ + h], ev);
}

// ---------------- propagate 1: nodes -> hyperedges ----------------
// he[b,n1,h,:] += Binv[n1] * alpha[e,h] * xw[b,n0,h,:]
__global__ void k_prop1(const float* __restrict__ xw,
                        const int*   __restrict__ hidx,
                        const float* __restrict__ logits, // exp values
                        const float* __restrict__ sumb,
                        const float* __restrict__ Bn,
                        float* __restrict__ he)           // [B,NHE,H,DH]
{
    long long gid = (long long)blockIdx.x * blockDim.x + threadIdx.x; // B*E*H*(DH/4)
    if (gid >= (long long)B_ * E_ * H_ * (DH / 4)) return;
    int d4 = gid & (DH / 4 - 1);
    int h  = (gid >> 5) & (H_ - 1);
    int e  = (int)((gid >> 7) & (E_ - 1));
    int b  = (int)(gid >> (7 + 14));                      // /(32*4*16384)
    int n0 = hidx[(b * 2 + 0) * E_ + e];
    int n1 = hidx[(b * 2 + 1) * E_ + e];

    float ev = logits[((b * E_ + e) * H_) + h];
    float s  = sumb[(b * L_ + n0) * H_ + h];
    float alpha = ev / (s + 1e-16f);
    float bn = Bn[b * NHE + n1];
    float binv = (bn > 0.0f) ? 1.0f / bn : 0.0f;
    float c = binv * alpha;

    const float4 xv = *reinterpret_cast<const float4*>(
        xw + (size_t)(b * L_ + n0) * C_ + h * DH + d4 * 4);
    float* dst = he + (size_t)(b * NHE + n1) * C_ + h * DH + d4 * 4;
    atomicAdd(dst + 0, c * xv.x);
    atomicAdd(dst + 1, c * xv.y);
    atomicAdd(dst + 2, c * xv.z);
    atomicAdd(dst + 3, c * xv.w);
}

// ---------------- propagate 2: hyperedges -> nodes ----------------
// out[b,n0,h,:] += Dinv[n0] * alpha[e,h] * he[b,n1,h,:]
__global__ void k_prop2(const int*   __restrict__ hidx,
                        const float* __restrict__ logits,
                        const float* __restrict__ sumb,
                        const float* __restrict__ Dn,
                        const float* __restrict__ he,
                        float* __restrict__ out)          // [B,L,C]
{
    long long gid = (long long)blockIdx.x * blockDim.x + threadIdx.x;
    if (gid >= (long long)B_ * E_ * H_ * (DH / 4)) return;
    int d4 = gid & (DH / 4 - 1);
    int h  = (gid >> 5) & (H_ - 1);
    int e  = (int)((gid >> 7) & (E_ - 1));
    int b  = (int)(gid >> (7 + 14));
    int n0 = hidx[(b * 2 + 0) * E_ + e];
    int n1 = hidx[(b * 2 + 1) * E_ + e];

    float ev = logits[((b * E_ + e) * H_) + h];
    float s  = sumb[(b * L_ + n0) * H_ + h];
    float alpha = ev / (s + 1e-16f);
    float dn = Dn[b * L_ + n0];
    float dinv = (dn > 0.0f) ? 1.0f / dn : 0.0f;
    float c = dinv * alpha;

    const float4 hv = *reinterpret_cast<const float4*>(
        he + (size_t)(b * NHE + n1) * C_ + h * DH + d4 * 4);
    float* dst = out + (size_t)(b * L_ + n0) * C_ + h * DH + d4 * 4;
    atomicAdd(dst + 0, c * hv.x);
    atomicAdd(dst + 1, c * hv.y);
    atomicAdd(dst + 2, c * hv.z);
    atomicAdd(dst + 3, c * hv.w);
}

// ---------------- launcher ----------------
extern "C" void kernel_launch(void* const* d_in, const int* in_sizes, int n_in,
                              void* d_out, int out_size, void* d_ws, size_t ws_size,
                              hipStream_t stream) {
    const float* x    = (const float*)d_in[0];    // [B,L,C]
    const int*   hidx = (const int*)d_in[1];      // [B,2,E]
    // d_in[2] = adj (unused by reference math)
    const float* W    = (const float*)d_in[3];    // [C,C]
    const float* att  = (const float*)d_in[4];    // [1,H,2*DH]
    float*       out  = (float*)d_out;            // [B,L,C]

    // workspace carve-up (floats)
    float* ws = (float*)d_ws;
    size_t off = 0;
    float* xw     = ws + off; off += (size_t)B_ * L_ * C_;   // 16,777,216
    float* he     = ws + off; off += (size_t)B_ * NHE * C_;  //  4,194,304
    float* logits = ws + off; off += (size_t)B_ * E_ * H_;   //  1,048,576
    float* sumb   = ws + off; off += (size_t)B_ * L_ * H_;   //    131,072
    float* Dn     = ws + off; off += (size_t)B_ * L_;        //     32,768
    float* Bn     = ws + off; off += (size_t)B_ * NHE;       //      8,192
    unsigned* maxb = (unsigned*)(ws + off);                  //    131,072 u32

    // zero accumulation buffers: he..Bn are contiguous
    {
        long long nz = (long long)B_ * NHE * C_ + (long long)B_ * E_ * H_
                     + (long long)B_ * L_ * H_ + (long long)B_ * L_ + (long long)B_ * NHE;
        k_zero_f32<<<2048, 256, 0, stream>>>(he, nz);
        k_zero_f32<<<2048, 256, 0, stream>>>(out, (long long)B_ * L_ * C_);
        k_init_max<<<512, 256, 0, stream>>>(maxb, (long long)B_ * L_ * H_);
    }

    // GEMM: grid = (C/BN, L/BM, B)
    {
        dim3 grid(C_ / BN, L_ / BM, B_);
        k_gemm_wmma<<<grid, 256, 0, stream>>>(x, W, xw);
    }

    // degrees
    k_degree<<<(B_ * E_ + 255) / 256, 256, 0, stream>>>(hidx, Dn, Bn);

    // attention pipeline
    const int nBEH = B_ * E_ * H_;
    k_logits<<<(nBEH + 255) / 256, 256, 0, stream>>>(xw, hidx, att, logits, maxb);
    k_expsum<<<(nBEH + 255) / 256, 256, 0, stream>>>(hidx, logits, maxb, sumb);

    // propagations
    const long long nProp = (long long)B_ * E_ * H_ * (DH / 4);
    const int blocks = (int)((nProp + 255) / 256);
    k_prop1<<<blocks, 256, 0, stream>>>(xw, hidx, logits, sumb, Bn, he);
    k_prop2<<<blocks, 256, 0, stream>>>(hidx, logits, sumb, Dn, he, out);
}